// SSIM3D_79362405695909
// MI455X (gfx1250) — compile-verified
//
#include <hip/hip_runtime.h>

// CDNA5 / gfx1250 SSIM3D structure-term kernel.
// Box filter along T implemented as a banded matmul on V_WMMA_F32_16X16X4_F32.

typedef __attribute__((ext_vector_type(2))) float v2f;
typedef __attribute__((ext_vector_type(8))) float v8f;

#define T_DIM    20
#define HW       65536          // 256*256, power of two
#define HALF_WIN 5              // window 11
#define FSCALE   (1.0f / 11.0f) // 1/(C*ws), C=1
#define C2H      0.00045f       // (0.03^2)/2
#define SEPS     1e-8f
#define N_TOTAL  20971520.0f    // 16*20*1*256*256

__global__ void __launch_bounds__(256)
ssim3d_wmma_main(const float* __restrict__ img1,
                 const float* __restrict__ img2,
                 float* __restrict__ partials)
{
    const int lane  = threadIdx.x & 31;
    const int wave  = threadIdx.x >> 5;
    const int group = blockIdx.x * 8 + wave;   // 16-pixel group over B*H*W
    const int nidx  = lane & 15;               // A row (pixel) / B column (t_out)
    const int half  = lane >> 4;               // lane half selects K sub-pair

    // This lane's pixel for A-matrix rows: flat pixel p over B*H*W.
    const int pix = group * 16 + nidx;
    const int b   = pix >> 16;                 // / HW
    const int sp  = pix & (HW - 1);
    const size_t base = (size_t)b * ((size_t)T_DIM * HW) + (size_t)(2 * half) * HW + (size_t)sp;
    const float* p1 = img1 + base;
    const float* p2 = img2 + base;

    // Load this lane's 10 t-samples of each image: t = 4c + 2*half + j.
    // Each element of the 168MB inputs is read exactly once, coalesced.
    float xv[10], yv[10];
#pragma unroll
    for (int c = 0; c < 5; ++c) {
#pragma unroll
        for (int j = 0; j < 2; ++j) {
            float a  = p1[(size_t)(4 * c + j) * HW];
            float bb = p2[(size_t)(4 * c + j) * HW];
            xv[2 * c + j] = fminf(fmaxf(a, 0.0f), 1.0f);   // clip [0,1]
            yv[2 * c + j] = fminf(fmaxf(bb, 0.0f), 1.0f);
        }
    }

    float lsum = 0.0f;

    // Two N-tiles cover t_out = 0..15 and 16..19 (columns >= 20 are zeroed in B
    // and masked out of the reduction; EXEC stays all-ones around every WMMA).
#pragma unroll
    for (int nt = 0; nt < 2; ++nt) {
        const int tout = nt * 16 + nidx;       // this lane's output-t column
        v8f acc_x = {}, acc_y = {}, acc_xx = {}, acc_yy = {}, acc_xy = {};
#pragma unroll
        for (int c = 0; c < 5; ++c) {
            const float x0 = xv[2 * c], x1 = xv[2 * c + 1];
            const float y0 = yv[2 * c], y1 = yv[2 * c + 1];
            v2f Ax  = {x0, x1};
            v2f Ay  = {y0, y1};
            v2f Axx = {x0 * x0, x1 * x1};
            v2f Ayy = {y0 * y0, y1 * y1};
            v2f Axy = {x0 * y0, x1 * y1};
            // Constant banded filter matrix B[k][n] built from lane id (no memory).
            const int tin0 = 4 * c + 2 * half;
            const int tin1 = tin0 + 1;
            const float b0 = ((tout < T_DIM) && (tout - tin0 <= HALF_WIN) &&
                              (tin0 - tout <= HALF_WIN)) ? FSCALE : 0.0f;
            const float b1 = ((tout < T_DIM) && (tout - tin1 <= HALF_WIN) &&
                              (tin1 - tout <= HALF_WIN)) ? FSCALE : 0.0f;
            v2f Bf = {b0, b1};
            acc_x  = __builtin_amdgcn_wmma_f32_16x16x4_f32(false, Ax,  false, Bf, (short)0, acc_x,  false, false);
            acc_y  = __builtin_amdgcn_wmma_f32_16x16x4_f32(false, Ay,  false, Bf, (short)0, acc_y,  false, false);
            acc_xx = __builtin_amdgcn_wmma_f32_16x16x4_f32(false, Axx, false, Bf, (short)0, acc_xx, false, false);
            acc_yy = __builtin_amdgcn_wmma_f32_16x16x4_f32(false, Ayy, false, Bf, (short)0, acc_yy, false, false);
            acc_xy = __builtin_amdgcn_wmma_f32_16x16x4_f32(false, Axy, false, Bf, (short)0, acc_xy, false, false);
        }
        // Element-wise structure term on the D accumulators (all share one layout).
        const float validf = (tout < T_DIM) ? 1.0f : 0.0f;
#pragma unroll
        for (int r = 0; r < 8; ++r) {
            float mu1  = acc_x[r], mu2 = acc_y[r];
            float s1sq = acc_xx[r] - mu1 * mu1;
            float s2sq = acc_yy[r] - mu2 * mu2;
            float s12  = acc_xy[r] - mu1 * mu2;
            float s1 = sqrtf(fmaxf(s1sq, SEPS));
            float s2 = sqrtf(fmaxf(s2sq, SEPS));
            lsum += validf * (s12 + C2H) / (s1 * s2 + C2H + SEPS);
        }
    }

    // Deterministic wave32 reduction.
#pragma unroll
    for (int m = 16; m >= 1; m >>= 1)
        lsum += __shfl_xor(lsum, m, 32);

    __shared__ float wsum[8];
    if (lane == 0) wsum[wave] = lsum;
    __syncthreads();
    if (threadIdx.x == 0) {
        float t = 0.0f;
#pragma unroll
        for (int w = 0; w < 8; ++w) t += wsum[w];
        partials[blockIdx.x] = t;
    }
}

__global__ void __launch_bounds__(256)
ssim3d_final_reduce(const float* __restrict__ partials, int n,
                    float* __restrict__ out)
{
    __shared__ float s[256];
    float acc = 0.0f;
    for (int i = threadIdx.x; i < n; i += 256) acc += partials[i];
    s[threadIdx.x] = acc;
    __syncthreads();
#pragma unroll
    for (int off = 128; off > 0; off >>= 1) {
        if ((int)threadIdx.x < off) s[threadIdx.x] += s[threadIdx.x + off];
        __syncthreads();
    }
    if (threadIdx.x == 0) out[0] = s[0] * (1.0f / N_TOTAL);
}

extern "C" void kernel_launch(void* const* d_in, const int* in_sizes, int n_in,
                              void* d_out, int out_size, void* d_ws, size_t ws_size,
                              hipStream_t stream) {
    (void)in_sizes; (void)n_in; (void)out_size; (void)ws_size;
    const float* img1 = (const float*)d_in[0];
    const float* img2 = (const float*)d_in[1];
    float* out      = (float*)d_out;
    float* partials = (float*)d_ws;            // 8192 floats (32 KB) of scratch

    const int nGroups = (16 * 256 * 256) / 16; // B*H*W / 16 = 65536
    const int nBlocks = nGroups / 8;           // 8 waves per block -> 8192

    ssim3d_wmma_main<<<nBlocks, 256, 0, stream>>>(img1, img2, partials);
    ssim3d_final_reduce<<<1, 256, 0, stream>>>(partials, nBlocks, out);
}